// PointNet2_36515811950618
// MI455X (gfx1250) — compile-verified
//
#include <hip/hip_runtime.h>

typedef __attribute__((ext_vector_type(2))) float v2f;
typedef __attribute__((ext_vector_type(8))) float v8f;

#define DEV __device__ __forceinline__

// ---------------------------------------------------------------------------
// fp32 WMMA 16x16 tile: Y[mb:mb+16, nb:nb+16] = act(X @ W^T + bias)
// A frag (16x4): lane l<16 holds (M=l, K=k+{0,1}); lane l+16 holds (M=l, K=k+{2,3})
// B frag (4x16): lane l<16 holds (K=k+{0,1}, N=l); lane l+16 holds (K=k+{2,3}, N=l)
// C/D (16x16): vgpr r, lane l<16 -> (M=r, N=l); lane l+16 -> (M=r+8, N=l)
// W is row-major (Cout x Kp) so B[k][n] = W[n][k].
// ---------------------------------------------------------------------------
DEV void wmma_tile(const float* __restrict__ W, const float* __restrict__ bias,
                   const float* X, int ldx, float* Y, int ldy,
                   int mb, int nb, int Kp, int lane, int relu)
{
  int half = lane >> 4;
  int l = lane & 15;
  const float* xrow = X + (size_t)(mb + l) * ldx + half * 2;
  const float* wrow = W + (size_t)(nb + l) * Kp + half * 2;
  v8f acc = {0.f, 0.f, 0.f, 0.f, 0.f, 0.f, 0.f, 0.f};
  for (int k = 0; k < Kp; k += 4) {
    v2f a, b;
    a.x = xrow[k];     a.y = xrow[k + 1];
    b.x = wrow[k];     b.y = wrow[k + 1];
    acc = __builtin_amdgcn_wmma_f32_16x16x4_f32(false, a, false, b,
                                                (short)0, acc, false, false);
  }
  float bc = bias[nb + l];
#pragma unroll
  for (int r = 0; r < 8; ++r) {
    int row = mb + r + half * 8;
    float v = acc[r] + bc;
    if (relu) v = fmaxf(v, 0.f);
    Y[(size_t)row * ldy + nb + l] = v;
  }
}

DEV void wmma_layer(const float* __restrict__ W, const float* __restrict__ bias,
                    const float* X, int ldx, float* Y, int ldy,
                    int M, int Kp, int Cout, int lane, int wave, int nwaves, int relu)
{
  int mt = M >> 4, nt = Cout >> 4;
  int ntiles = mt * nt;
  for (int t = wave; t < ntiles; t += nwaves)
    wmma_tile(W, bias, X, ldx, Y, ldy, (t % mt) << 4, (t / mt) << 4, Kp, lane, relu);
}

// ---------------------------------------------------------------------------
// BN fold: wout[o][k] = w[o][k] * g[o] * inv_std (K padded with zeros),
//          bout[o]    = b[o] * g[o] * inv_std + be[o]
// ---------------------------------------------------------------------------
__global__ void fold_kernel(const float* __restrict__ w, const float* __restrict__ b,
                            const float* __restrict__ g, const float* __restrict__ be,
                            int Cout, int Cin, int Kp,
                            float* __restrict__ wout, float* __restrict__ bout)
{
  const float inv_std = 0.999995000037f; // 1/sqrt(1 + 1e-5)
  int e = blockIdx.x * blockDim.x + threadIdx.x;
  if (e < Cout * Kp) {
    int o = e / Kp, k = e - o * Kp;
    wout[e] = (k < Cin) ? w[o * Cin + k] * g[o] * inv_std : 0.0f;
  }
  if (e < Cout) bout[e] = b[e] * g[e] * inv_std + be[e];
}

// ---------------------------------------------------------------------------
// Farthest point sampling: one block per batch, xyz + dist LDS-resident.
// Dynamic LDS: 4*N floats (x, y, z, dist).
// ---------------------------------------------------------------------------
__global__ void fps_kernel(const float* __restrict__ xyz, int N, int S,
                           float* __restrict__ newxyz, float* __restrict__ outT)
{
  extern __shared__ float sm[];
  float* px = sm;
  float* py = sm + N;
  float* pz = sm + 2 * N;
  float* dist = sm + 3 * N;
  __shared__ float rv[8];
  __shared__ int   ri[8];
  __shared__ int   s_far;

  int b = blockIdx.x, tid = threadIdx.x;
  int lane = tid & 31, wave = tid >> 5;
  int nw = blockDim.x >> 5;
  const float* base = xyz + (size_t)b * N * 3;

  for (int i = tid; i < N; i += blockDim.x) {
    px[i] = base[i * 3 + 0];
    py[i] = base[i * 3 + 1];
    pz[i] = base[i * 3 + 2];
    dist[i] = 1e10f;
  }
  if (tid == 0) s_far = 0;
  __syncthreads();

  for (int s = 0; s < S; ++s) {
    int fi = s_far;
    float cx = px[fi], cy = py[fi], cz = pz[fi];
    if (tid == 0) {
      size_t o = ((size_t)b * S + s) * 3;
      newxyz[o + 0] = cx; newxyz[o + 1] = cy; newxyz[o + 2] = cz;
      outT[((size_t)b * 3 + 0) * S + s] = cx;
      outT[((size_t)b * 3 + 1) * S + s] = cy;
      outT[((size_t)b * 3 + 2) * S + s] = cz;
    }
    float bv = -1.f; int bi = 0;
    for (int i = tid; i < N; i += blockDim.x) {
      float dx = px[i] - cx, dy = py[i] - cy, dz = pz[i] - cz;
      float d = dx * dx + dy * dy + dz * dz;
      float nd = fminf(dist[i], d);
      dist[i] = nd;
      if (nd > bv) { bv = nd; bi = i; }
    }
    for (int off = 16; off > 0; off >>= 1) {
      float ov = __shfl_xor(bv, off, 32);
      int   oi = __shfl_xor(bi, off, 32);
      if (ov > bv || (ov == bv && oi < bi)) { bv = ov; bi = oi; }
    }
    if (lane == 0) { rv[wave] = bv; ri[wave] = bi; }
    __syncthreads();
    if (tid == 0) {
      float bbv = rv[0]; int bbi = ri[0];
      for (int w = 1; w < nw; ++w)
        if (rv[w] > bbv || (rv[w] == bbv && ri[w] < bbi)) { bbv = rv[w]; bbi = ri[w]; }
      s_far = bbi;
    }
    __syncthreads();
  }
}

// ---------------------------------------------------------------------------
// Ball query: one wave per center; ballot-based ordered compaction.
// ---------------------------------------------------------------------------
__global__ void ballquery_kernel(const float* __restrict__ xyz, int N, int S,
                                 const float* __restrict__ centers, int nc,
                                 float r2, int ns, int* __restrict__ out)
{
  int lane = threadIdx.x & 31;
  int wid = threadIdx.x >> 5;
  int c = blockIdx.x * (blockDim.x >> 5) + wid;
  if (c >= nc) return;
  float cx = centers[(size_t)c * 3 + 0];
  float cy = centers[(size_t)c * 3 + 1];
  float cz = centers[(size_t)c * 3 + 2];
  const float* base = xyz + (size_t)(c / S) * N * 3;
  int count = 0, firstIdx = -1;
  for (int n0 = 0; n0 < N && count < ns; n0 += 32) {
    int i = n0 + lane;
    float dx = base[i * 3 + 0] - cx;
    float dy = base[i * 3 + 1] - cy;
    float dz = base[i * 3 + 2] - cz;
    bool inR = (dx * dx + dy * dy + dz * dz) <= r2;
    unsigned mask = (unsigned)__ballot(inR);
    if (firstIdx < 0 && mask) firstIdx = n0 + __ffs(mask) - 1;
    int pre = __popc(mask & ((1u << lane) - 1u));
    if (inR && (count + pre) < ns) out[(size_t)c * ns + count + pre] = i;
    count += __popc(mask);
  }
  if (count < ns) {
    if (firstIdx < 0) firstIdx = 0;
    for (int j = count + lane; j < ns; j += 32) out[(size_t)c * ns + j] = firstIdx;
  }
}

// ---------------------------------------------------------------------------
// SA1: one block (4 waves) per center. 32 samples, 3->64->64->128, max over 32.
// ---------------------------------------------------------------------------
__global__ void __launch_bounds__(128) sa1_kernel(
    const float* __restrict__ xyz, const float* __restrict__ cent,
    const int* __restrict__ idx,
    const float* __restrict__ W1, const float* __restrict__ B1,
    const float* __restrict__ W2, const float* __restrict__ B2,
    const float* __restrict__ W3, const float* __restrict__ B3,
    float* __restrict__ outp)
{
  const int N = 8192, K = 32;
  __shared__ float X0[32 * 4];
  __shared__ float Y1[32 * 64];
  __shared__ float Y2[32 * 64];
  __shared__ float Y3[32 * 128];
  __shared__ int sidx[32];
  int g = blockIdx.x;          // b*512 + s
  int b = g >> 9;
  int tid = threadIdx.x, lane = tid & 31, wave = tid >> 5;
  float cx = cent[(size_t)g * 3 + 0];
  float cy = cent[(size_t)g * 3 + 1];
  float cz = cent[(size_t)g * 3 + 2];
  if (tid < K) sidx[tid] = idx[(size_t)g * K + tid];
  __syncthreads();
  if (tid < K) {
    const float* p = xyz + ((size_t)b * N + sidx[tid]) * 3;
    X0[tid * 4 + 0] = p[0] - cx;
    X0[tid * 4 + 1] = p[1] - cy;
    X0[tid * 4 + 2] = p[2] - cz;
    X0[tid * 4 + 3] = 0.f;
  }
  __syncthreads();
  wmma_layer(W1, B1, X0, 4,  Y1, 64,  32, 4,  64,  lane, wave, 4, 1); __syncthreads();
  wmma_layer(W2, B2, Y1, 64, Y2, 64,  32, 64, 64,  lane, wave, 4, 1); __syncthreads();
  wmma_layer(W3, B3, Y2, 64, Y3, 128, 32, 64, 128, lane, wave, 4, 1); __syncthreads();
  float m = Y3[tid];
  for (int r = 1; r < K; ++r) m = fmaxf(m, Y3[r * 128 + tid]);
  outp[(size_t)g * 128 + tid] = m;
}

// ---------------------------------------------------------------------------
// SA2: one block (8 waves) per center. 64 samples, (3+128)->128->128->256.
// Dynamic LDS: 33024 floats (A:64x132 then Y2 | Y1:64x128 | Y3:64x256).
// ---------------------------------------------------------------------------
__global__ void __launch_bounds__(256) sa2_kernel(
    const float* __restrict__ xyz1, const float* __restrict__ cent,
    const int* __restrict__ idx, const float* __restrict__ l1p,
    const float* __restrict__ W1, const float* __restrict__ B1,
    const float* __restrict__ W2, const float* __restrict__ B2,
    const float* __restrict__ W3, const float* __restrict__ B3,
    float* __restrict__ outp)
{
  const int N = 512, K = 64, KP0 = 132;
  extern __shared__ float sm[];
  float* A  = sm;                    // 64*132, later reused for Y2 (64*128)
  float* Bf = sm + 64 * 132;         // Y1: 64*128
  float* Cf = Bf + 64 * 128;         // Y3: 64*256
  __shared__ int sidx[64];
  int g = blockIdx.x;                // b*128 + s
  int b = g >> 7;
  int tid = threadIdx.x, lane = tid & 31, wave = tid >> 5;
  float cx = cent[(size_t)g * 3 + 0];
  float cy = cent[(size_t)g * 3 + 1];
  float cz = cent[(size_t)g * 3 + 2];
  if (tid < K) sidx[tid] = idx[(size_t)g * K + tid];
  __syncthreads();
  for (int e = tid; e < K * KP0; e += 256) {
    int r = e / KP0, c = e - r * KP0;
    int pi = sidx[r];
    float v;
    if (c < 3)
      v = xyz1[((size_t)b * N + pi) * 3 + c] - (c == 0 ? cx : (c == 1 ? cy : cz));
    else if (c < 131)
      v = l1p[((size_t)b * N + pi) * 128 + (c - 3)];
    else
      v = 0.f;
    A[e] = v;
  }
  __syncthreads();
  wmma_layer(W1, B1, A, 132,  Bf, 128, 64, 132, 128, lane, wave, 8, 1); __syncthreads();
  wmma_layer(W2, B2, Bf, 128, A,  128, 64, 128, 128, lane, wave, 8, 1); __syncthreads();
  wmma_layer(W3, B3, A, 128,  Cf, 256, 64, 128, 256, lane, wave, 8, 1); __syncthreads();
  float m = Cf[tid];
  for (int r = 1; r < K; ++r) m = fmaxf(m, Cf[r * 256 + tid]);
  outp[(size_t)g * 256 + tid] = m;
}

// ---------------------------------------------------------------------------
// SA3 input: X (2048 x 260) = [l2_xyz(3) | l2_points(256) | pad(1)]
// ---------------------------------------------------------------------------
__global__ void build_l3_input(const float* __restrict__ l2xyz,
                               const float* __restrict__ l2p,
                               float* __restrict__ X)
{
  const int ROWS = 2048, KP = 260;
  int e = blockIdx.x * blockDim.x + threadIdx.x;
  if (e >= ROWS * KP) return;
  int r = e / KP, c = e - r * KP;
  float v;
  if (c < 3)        v = l2xyz[(size_t)r * 3 + c];
  else if (c < 259) v = l2p[(size_t)r * 256 + (c - 3)];
  else              v = 0.f;
  X[e] = v;
}

// ---------------------------------------------------------------------------
// Generic global GEMM: wave-per-16x16-tile, Y = act(X @ W^T + bias)
// ---------------------------------------------------------------------------
__global__ void __launch_bounds__(256) gemm_kernel(
    const float* __restrict__ X, const float* __restrict__ W,
    const float* __restrict__ bias, float* __restrict__ Y,
    int M, int Kp, int N, int relu)
{
  int lane = threadIdx.x & 31, wave = threadIdx.x >> 5;
  int tile = blockIdx.x * (blockDim.x >> 5) + wave;
  int mt = M >> 4, nt = N >> 4;
  if (tile >= mt * nt) return;
  wmma_tile(W, bias, X, Kp, Y, N, (tile % mt) << 4, (tile / mt) << 4, Kp, lane, relu);
}

// max over 128 rows per batch: (16,128,1024) -> (16,1024)
__global__ void max_rows_kernel(const float* __restrict__ Y, float* __restrict__ out)
{
  int e = blockIdx.x * blockDim.x + threadIdx.x;
  if (e >= 16 * 1024) return;
  int b = e >> 10, c = e & 1023;
  const float* p = Y + (size_t)b * 128 * 1024 + c;
  float m = p[0];
  for (int r = 1; r < 128; ++r) m = fmaxf(m, p[(size_t)r * 1024]);
  out[e] = m;
}

__global__ void zero_kernel(float* __restrict__ p, int n)
{
  int i = blockIdx.x * blockDim.x + threadIdx.x;
  if (i < n) p[i] = 0.f;
}

// ---------------------------------------------------------------------------
extern "C" void kernel_launch(void* const* d_in, const int* in_sizes, int n_in,
                              void* d_out, int out_size, void* d_ws, size_t ws_size,
                              hipStream_t stream)
{
  (void)in_sizes; (void)n_in; (void)out_size; (void)ws_size;
  const float* xyz    = (const float*)d_in[0];
  const float* proj_w = (const float*)d_in[37];
  const float* proj_b = (const float*)d_in[38];
  float* ws   = (float*)d_ws;
  float* outp = (float*)d_out;

  // workspace layout (float offsets)
  float* newxyz1 = ws;                        // 16*512*3      = 24576
  int*   idx1    = (int*)(ws + 24576);        // 16*512*32     = 262144 ints
  float* l1p     = ws + 286720;               // 16*512*128    = 1048576
  float* newxyz2 = ws + 1335296;              // 16*128*3      = 6144
  int*   idx2    = (int*)(ws + 1341440);      // 16*128*64     = 131072 ints
  float* l2p     = ws + 1472512;              // 16*128*256    = 524288
  float* X3      = ws + 1996800;              // 2048*260      = 532480
  float* Y3a     = ws + 2529280;              // 2048*256      = 524288
  float* Y3b     = ws + 3053568;              // 2048*512      = 1048576
  float* Y3c     = ws + 4102144;              // 2048*1024     = 2097152
  float* l3f     = ws + 6199296;              // 16*1024       = 16384
  float* wf      = ws + 6215680;              // folded weights

  struct LD { int co, ci, kp; };
  const LD dims[9] = {{64, 3, 4},    {64, 64, 64},   {128, 64, 64},
                      {128, 131, 132},{128, 128, 128},{256, 128, 128},
                      {256, 259, 260},{512, 256, 256},{1024, 512, 512}};
  float* Wf[9]; float* Bb[9];
  size_t o = 0;
  for (int i = 0; i < 9; ++i) {
    Wf[i] = wf + o; o += (size_t)dims[i].co * dims[i].kp;
    Bb[i] = wf + o; o += dims[i].co;
  }
  for (int i = 0; i < 9; ++i) {
    const float* w  = (const float*)d_in[1 + i * 4 + 0];
    const float* bi = (const float*)d_in[1 + i * 4 + 1];
    const float* gg = (const float*)d_in[1 + i * 4 + 2];
    const float* be = (const float*)d_in[1 + i * 4 + 3];
    int n = dims[i].co * dims[i].kp;
    fold_kernel<<<(n + 255) / 256, 256, 0, stream>>>(w, bi, gg, be,
        dims[i].co, dims[i].ci, dims[i].kp, Wf[i], Bb[i]);
  }

  // output regions (concatenated in return order)
  float* out_l3pts  = outp;          // (16,1,256)   = 4096
  float* out_l3xyzT = outp + 4096;   // (16,3,1)     = 48
  float* out_l2xyzT = outp + 4144;   // (16,3,128)   = 6144
  float* out_l1xyzT = outp + 10288;  // (16,3,512)   = 24576

  zero_kernel<<<1, 64, 0, stream>>>(out_l3xyzT, 48);

  // ---- SA1 ----
  fps_kernel<<<16, 256, 4 * 8192 * sizeof(float), stream>>>(
      xyz, 8192, 512, newxyz1, out_l1xyzT);
  ballquery_kernel<<<8192 / 8, 256, 0, stream>>>(
      xyz, 8192, 512, newxyz1, 8192, 0.04f, 32, idx1);
  sa1_kernel<<<8192, 128, 0, stream>>>(xyz, newxyz1, idx1,
      Wf[0], Bb[0], Wf[1], Bb[1], Wf[2], Bb[2], l1p);

  // ---- SA2 ----
  fps_kernel<<<16, 256, 4 * 512 * sizeof(float), stream>>>(
      newxyz1, 512, 128, newxyz2, out_l2xyzT);
  ballquery_kernel<<<2048 / 8, 256, 0, stream>>>(
      newxyz1, 512, 128, newxyz2, 2048, 0.16f, 64, idx2);
  sa2_kernel<<<2048, 256, 33024 * sizeof(float), stream>>>(
      newxyz1, newxyz2, idx2, l1p,
      Wf[3], Bb[3], Wf[4], Bb[4], Wf[5], Bb[5], l2p);

  // ---- SA3 (group all) ----
  build_l3_input<<<(2048 * 260 + 255) / 256, 256, 0, stream>>>(newxyz2, l2p, X3);
  gemm_kernel<<<(128 * 16) / 8, 256, 0, stream>>>(X3,  Wf[6], Bb[6], Y3a, 2048, 260, 256,  1);
  gemm_kernel<<<(128 * 32) / 8, 256, 0, stream>>>(Y3a, Wf[7], Bb[7], Y3b, 2048, 256, 512,  1);
  gemm_kernel<<<(128 * 64) / 8, 256, 0, stream>>>(Y3b, Wf[8], Bb[8], Y3c, 2048, 512, 1024, 1);
  max_rows_kernel<<<(16 * 1024 + 255) / 256, 256, 0, stream>>>(Y3c, l3f);

  // ---- projection: (16,1024) @ (256,1024)^T + b ----
  gemm_kernel<<<2, 256, 0, stream>>>(l3f, proj_w, proj_b, out_l3pts, 16, 1024, 256, 0);
}